// GraphDQN_18915035971935
// MI455X (gfx1250) — compile-verified
//
#include <hip/hip_runtime.h>
#include <hip/hip_bf16.h>

// ---------------------------------------------------------------------------
// GraphDQN pipeline for gfx1250 (MI455X).
//  - conv2/conv3: implicit GEMM on V_WMMA_F32_16X16X4_F32, A-tile im2col-staged
//    in LDS (no per-step gathers / div-mod in the MMA loop).
//  - Bellman-Ford min-plus: per-batch 256x256 cost matrix resident in the
//    320KB WGP LDS, staged with GLOBAL_LOAD_ASYNC_TO_LDS_B128.
// ---------------------------------------------------------------------------

typedef float v2f __attribute__((ext_vector_type(2)));
typedef float v8f __attribute__((ext_vector_type(8)));
typedef int v4i __attribute__((vector_size(16)));  // async-LDS builtin element type

#if defined(__gfx1250__) &&                                                    \
    __has_builtin(__builtin_amdgcn_global_load_async_to_lds_b128) &&           \
    __has_builtin(__builtin_amdgcn_s_wait_asynccnt)
#define HAVE_ASYNC_LDS 1
#else
#define HAVE_ASYNC_LDS 0
#endif

__device__ __forceinline__ v8f wmma_f32_16x16x4(v2f a, v2f b, v8f c) {
  // (neg_a, A, neg_b, B, c_mod, C, reuse_a, reuse_b)
  return __builtin_amdgcn_wmma_f32_16x16x4_f32(false, a, false, b, (short)0, c,
                                               false, false);
}

__device__ __forceinline__ unsigned long long umin64(unsigned long long a,
                                                     unsigned long long b) {
  return a < b ? a : b;
}

// ---------------------------------------------------------------------------
// Stage 1: target-color encoder -> seq row 0.  grid=64, block=64
// ---------------------------------------------------------------------------
__global__ void k_tenc(const float* __restrict__ x,
                       const float* __restrict__ w1, const float* __restrict__ b1,
                       const float* __restrict__ w2, const float* __restrict__ b2,
                       float* __restrict__ seq) {
  __shared__ float tb[64];
  int b = blockIdx.x, t = threadIdx.x;
  float x0 = x[(size_t)(b * 3 + 0) * 4096];
  float x1 = x[(size_t)(b * 3 + 1) * 4096];
  float x2 = x[(size_t)(b * 3 + 2) * 4096];
  tb[t] = fmaxf(x0 * w1[t * 3 + 0] + x1 * w1[t * 3 + 1] + x2 * w1[t * 3 + 2] + b1[t], 0.f);
  __syncthreads();
  if (t < 14) {
    float s = b2[t];
    for (int i = 0; i < 64; ++i) s += tb[i] * w2[t * 64 + i];
    seq[b * 70 + t] = s;
  }
}

// ---------------------------------------------------------------------------
// Stage 2: conv1(3x3 VALID) + avgpool2 + relu, fused.  984064 threads.
// h1: (B,16,31,31)
// ---------------------------------------------------------------------------
__global__ void k_conv1(const float* __restrict__ x, const float* __restrict__ w,
                        const float* __restrict__ bias, float* __restrict__ h1) {
  int idx = blockIdx.x * 256 + threadIdx.x;  // == 64*16*961 total
  int m = idx % 961;
  int r = idx / 961;
  int oc = r & 15, b = r >> 4;
  int oy = m / 31, ox = m % 31;
  const float* xb = x + (size_t)b * 3 * 4096;
  const float* wf = w + oc * 27;
  float acc = 0.f;
  for (int py = 0; py < 2; ++py)
    for (int px = 0; px < 2; ++px) {
      int cy = 2 * oy + py, cx = 2 * ox + px;
      for (int ic = 0; ic < 3; ++ic)
        #pragma unroll
        for (int ky = 0; ky < 3; ++ky)
          #pragma unroll
          for (int kx = 0; kx < 3; ++kx)
            acc += (xb[ic * 4096 + (cy + ky) * 64 + (cx + kx)] - 0.5f) *
                   wf[ic * 9 + ky * 3 + kx];
    }
  h1[idx] = fmaxf(acc * 0.25f + bias[oc], 0.f);
}

// ---------------------------------------------------------------------------
// Stage 3: conv2 (pad 1, 16->32) implicit GEMM on WMMA f32 16x16x4.
// grid=(61 m-tiles, 64 batch), block=32 (one wave).
// A-tile (16 x 144) im2col-staged into LDS; MMA loop is pure LDS + WMMA.
// out: (B,32,961) pre-pool.
// ---------------------------------------------------------------------------
__global__ void k_conv2(const float* __restrict__ h1, const float* __restrict__ w2,
                        const float* __restrict__ bias, float* __restrict__ out) {
  __shared__ float ws[32 * 144];   // weights, (oc, k)
  __shared__ float as[16 * 144];   // im2col A-tile, (m-row, k)
  int b = blockIdx.y;
  int mbase = blockIdx.x * 16;
  int lane = threadIdx.x;
  // cooperative vectorized weight stage
  for (int i = lane * 4; i < 32 * 144; i += 32 * 4)
    *(float4*)&ws[i] = *(const float4*)&w2[i];
  int half = lane >> 4, lrow = lane & 15;
  int m = mbase + lrow;
  bool mval = m < 961;
  int oy = m / 31, ox = m % 31;
  const float* hb = h1 + (size_t)b * 16 * 961;
  // im2col stage: lane (lrow, half) stages ic in [half*8, half*8+8)
  for (int i8 = 0; i8 < 8; ++i8) {
    int ic = half * 8 + i8;
    const float* src = hb + ic * 961;
    #pragma unroll
    for (int ky = 0; ky < 3; ++ky) {
      int iy = oy + ky - 1;
      bool yok = mval && iy >= 0 && iy < 31;
      float v0 = 0.f, v1 = 0.f, v2 = 0.f;
      if (yok) {
        const float* rp = src + iy * 31 + ox;
        if (ox >= 1) v0 = rp[-1];
        v1 = rp[0];
        if (ox <= 29) v2 = rp[1];
      }
      int kb = ic * 9 + ky * 3;
      as[lrow * 144 + kb + 0] = v0;
      as[lrow * 144 + kb + 1] = v1;
      as[lrow * 144 + kb + 2] = v2;
    }
  }
  __syncthreads();
  v8f c0 = {}, c1 = {};
  #pragma unroll 4
  for (int ks = 0; ks < 36; ++ks) {
    int k0 = ks * 4 + half * 2;  // f32 A/B layout: lanes>=16 hold K+2
    v2f a, bA, bB;
    a[0]  = as[lrow * 144 + k0];        a[1]  = as[lrow * 144 + k0 + 1];
    bA[0] = ws[lrow * 144 + k0];        bA[1] = ws[lrow * 144 + k0 + 1];
    bB[0] = ws[(lrow + 16) * 144 + k0]; bB[1] = ws[(lrow + 16) * 144 + k0 + 1];
    c0 = wmma_f32_16x16x4(a, bA, c0);
    c1 = wmma_f32_16x16x4(a, bB, c1);
  }
  float b0 = bias[lrow], b1 = bias[lrow + 16];
  float* ob = out + (size_t)b * 32 * 961;
  #pragma unroll
  for (int i = 0; i < 8; ++i) {  // D: lane holds D[half*8+i][lrow]
    int mm = mbase + half * 8 + i;
    if (mm < 961) {
      ob[lrow * 961 + mm]        = c0[i] + b0;
      ob[(lrow + 16) * 961 + mm] = c1[i] + b1;
    }
  }
}

// ---------------------------------------------------------------------------
// Stage 4: avgpool2 + relu on conv2 output.  h2: (B,32,15,15)
// ---------------------------------------------------------------------------
__global__ void k_pool2(const float* __restrict__ cf, float* __restrict__ h2) {
  int idx = blockIdx.x * 256 + threadIdx.x;  // == 64*32*225 total
  int m = idx % 225;
  int r = idx / 225;
  int oc = r & 31, b = r >> 5;
  int py = m / 15, px = m % 15;
  const float* cb = cf + (size_t)(b * 32 + oc) * 961;
  float s = cb[(2 * py) * 31 + 2 * px] + cb[(2 * py) * 31 + 2 * px + 1] +
            cb[(2 * py + 1) * 31 + 2 * px] + cb[(2 * py + 1) * 31 + 2 * px + 1];
  h2[idx] = fmaxf(s * 0.25f, 0.f);
}

// ---------------------------------------------------------------------------
// Stage 5: conv3 (pad 1, 32->32) + relu, implicit GEMM WMMA.  grid=(15,64)x32
// h3: (B,32,225)
// ---------------------------------------------------------------------------
__global__ void k_conv3(const float* __restrict__ h2, const float* __restrict__ w3,
                        const float* __restrict__ bias, float* __restrict__ h3) {
  __shared__ float ws[32 * 288];   // 36 KB
  __shared__ float as[16 * 288];   // 18 KB
  int b = blockIdx.y;
  int mbase = blockIdx.x * 16;
  int lane = threadIdx.x;
  for (int i = lane * 4; i < 32 * 288; i += 32 * 4)
    *(float4*)&ws[i] = *(const float4*)&w3[i];
  int half = lane >> 4, lrow = lane & 15;
  int m = mbase + lrow;
  bool mval = m < 225;
  int oy = m / 15, ox = m % 15;
  const float* hb = h2 + (size_t)b * 32 * 225;
  for (int i16 = 0; i16 < 16; ++i16) {
    int ic = half * 16 + i16;
    const float* src = hb + ic * 225;
    #pragma unroll
    for (int ky = 0; ky < 3; ++ky) {
      int iy = oy + ky - 1;
      bool yok = mval && iy >= 0 && iy < 15;
      float v0 = 0.f, v1 = 0.f, v2 = 0.f;
      if (yok) {
        const float* rp = src + iy * 15 + ox;
        if (ox >= 1) v0 = rp[-1];
        v1 = rp[0];
        if (ox <= 13) v2 = rp[1];
      }
      int kb = ic * 9 + ky * 3;
      as[lrow * 288 + kb + 0] = v0;
      as[lrow * 288 + kb + 1] = v1;
      as[lrow * 288 + kb + 2] = v2;
    }
  }
  __syncthreads();
  v8f c0 = {}, c1 = {};
  #pragma unroll 4
  for (int ks = 0; ks < 72; ++ks) {
    int k0 = ks * 4 + half * 2;
    v2f a, bA, bB;
    a[0]  = as[lrow * 288 + k0];        a[1]  = as[lrow * 288 + k0 + 1];
    bA[0] = ws[lrow * 288 + k0];        bA[1] = ws[lrow * 288 + k0 + 1];
    bB[0] = ws[(lrow + 16) * 288 + k0]; bB[1] = ws[(lrow + 16) * 288 + k0 + 1];
    c0 = wmma_f32_16x16x4(a, bA, c0);
    c1 = wmma_f32_16x16x4(a, bB, c1);
  }
  float b0 = bias[lrow], b1 = bias[lrow + 16];
  float* ob = h3 + (size_t)b * 32 * 225;
  #pragma unroll
  for (int i = 0; i < 8; ++i) {
    int mm = mbase + half * 8 + i;
    if (mm < 225) {
      ob[lrow * 225 + mm]        = fmaxf(c0[i] + b0, 0.f);
      ob[(lrow + 16) * 225 + mm] = fmaxf(c1[i] + b1, 0.f);
    }
  }
}

// ---------------------------------------------------------------------------
// Stage 6: vision encoder: (B,7200) @ (7200,8)^T.  grid=(8,64), block=256
// ---------------------------------------------------------------------------
__global__ void k_venc(const float* __restrict__ h3, const float* __restrict__ w,
                       const float* __restrict__ bias, float* __restrict__ ve) {
  int f = blockIdx.x, b = blockIdx.y, t = threadIdx.x;
  const float* hh = h3 + (size_t)b * 7200;
  const float* wf = w + (size_t)f * 7200;
  float p = 0.f;
  for (int k = t; k < 7200; k += 256) p += hh[k] * wf[k];
  __shared__ float red[256];
  red[t] = p;
  __syncthreads();
  for (int off = 128; off; off >>= 1) {
    if (t < off) red[t] += red[t + off];
    __syncthreads();
  }
  if (t == 0) ve[b * 8 + f] = red[0] + bias[f];
}

// ---------------------------------------------------------------------------
// Stage 7: nearest memory node (argmin of L2, first-index tie-break)
// ---------------------------------------------------------------------------
__global__ void k_closest(const float* __restrict__ ve,
                          const float* __restrict__ nodes, int* __restrict__ cl) {
  int b = blockIdx.x, s = threadIdx.x;  // block 256
  float d2 = 0.f;
  #pragma unroll
  for (int f = 0; f < 8; ++f) {
    float df = ve[b * 8 + f] - nodes[((size_t)b * 256 + s) * 8 + f];
    d2 += df * df;
  }
  __shared__ unsigned long long red[256];
  red[s] = ((unsigned long long)__float_as_uint(d2) << 32) | (unsigned)s;
  __syncthreads();
  for (int off = 128; off; off >>= 1) {
    if (s < off) red[s] = umin64(red[s], red[s + off]);
    __syncthreads();
  }
  if (s == 0) cl[b] = (int)(red[0] & 0xffffffffu);
}

// ---------------------------------------------------------------------------
// Stage 8: cost = mem_edges.min(axis=-1).  3 aligned float2 loads per thread.
// ---------------------------------------------------------------------------
__global__ void k_cost(const float* __restrict__ me, float* __restrict__ cost) {
  size_t idx = (size_t)blockIdx.x * 256 + threadIdx.x;  // 64*256*256
  const float2* p = (const float2*)me + idx * 3;
  float2 a = p[0], b = p[1], c = p[2];
  float m = fminf(fminf(fminf(a.x, a.y), fminf(b.x, b.y)), fminf(c.x, c.y));
  cost[idx] = m;
}

// ---------------------------------------------------------------------------
// Stage 9: 255x min-plus relaxation; 256x256 cost matrix LDS-resident
// (257KB of the 320KB WGP LDS), staged via async global->LDS B128 copies.
// grid=64, block=256.
// ---------------------------------------------------------------------------
__global__ void k_bellman(const float* __restrict__ cost,
                          const int* __restrict__ cl, float* __restrict__ dfin) {
  __shared__ float cs[256 * 256];  // 256 KB
  __shared__ float dcur[256];
  int b = blockIdx.x, t = threadIdx.x;
  const float* cb = cost + (size_t)b * 65536;
#if HAVE_ASYNC_LDS
  {
    __attribute__((address_space(1))) v4i* gsrc =
        (__attribute__((address_space(1))) v4i*)(float*)cb;
    __attribute__((address_space(3))) v4i* ldst =
        (__attribute__((address_space(3))) v4i*)cs;
    for (int i = 0; i < 64; ++i) {
      int off = i * 256 + t;  // 16B vectors: 16384 total
      __builtin_amdgcn_global_load_async_to_lds_b128(gsrc + off, ldst + off, 0, 0);
    }
    __builtin_amdgcn_s_wait_asynccnt(0);
  }
#else
  for (int i = t; i < 65536; i += 256) cs[i] = cb[i];
#endif
  __syncthreads();
  int c0 = cl[b];
  float dv = cs[c0 * 256 + t];
  if (t == c0) dv = 0.f;
  dcur[t] = dv;
  __syncthreads();
  for (int it = 0; it < 255; ++it) {
    float m = dcur[t];
    for (int j = 0; j < 256; ++j) m = fminf(m, dcur[j] + cs[j * 256 + t]);
    __syncthreads();
    dcur[t] = m;
    __syncthreads();
  }
  dfin[b * 256 + t] = dcur[t];
}

// ---------------------------------------------------------------------------
// Stage 10: top-4 smallest path distances (stable, lower-index ties first).
// ---------------------------------------------------------------------------
__global__ void k_topk(const float* __restrict__ dfin, int* __restrict__ topk) {
  int b = blockIdx.x, t = threadIdx.x;  // block 256
  float v = dfin[b * 256 + t];
  if (isinf(v)) v = 1e9f;
  __shared__ unsigned long long red[256];
  for (int r = 0; r < 4; ++r) {
    red[t] = ((unsigned long long)__float_as_uint(v) << 32) | (unsigned)t;
    __syncthreads();
    for (int off = 128; off; off >>= 1) {
      if (t < off) red[t] = umin64(red[t], red[t + off]);
      __syncthreads();
    }
    int bi = (int)(red[0] & 0xffffffffu);
    if (t == 0) topk[b * 4 + r] = bi;
    if (t == bi) v = 3.4e38f;  // exclude from later rounds
    __syncthreads();
  }
}

// ---------------------------------------------------------------------------
// Stage 11: gather retrieved nodes + per-row min over outgoing edges.
// grid=(4,64), block=256.  seq rows 1..4.
// ---------------------------------------------------------------------------
__global__ void k_gather(const float* __restrict__ nodes,
                         const float* __restrict__ me,
                         const int* __restrict__ topk, float* __restrict__ seq) {
  int i = blockIdx.x, b = blockIdx.y, t = threadIdx.x;
  int idx = topk[b * 4 + i];
  const float* row = me + (((size_t)b * 256 + idx) * 256) * 6;  // [j][e]
  __shared__ float red[6 * 256];
  #pragma unroll
  for (int e = 0; e < 6; ++e) red[e * 256 + t] = row[t * 6 + e];
  __syncthreads();
  for (int off = 128; off; off >>= 1) {
    if (t < off)
      #pragma unroll
      for (int e = 0; e < 6; ++e)
        red[e * 256 + t] = fminf(red[e * 256 + t], red[e * 256 + t + off]);
    __syncthreads();
  }
  if (t < 6) seq[b * 70 + (1 + i) * 14 + 8 + t] = red[t * 256];
  if (t >= 32 && t < 40) {
    int f = t - 32;
    seq[b * 70 + (1 + i) * 14 + f] = nodes[((size_t)b * 256 + idx) * 8 + f];
  }
}

// ---------------------------------------------------------------------------
// Stage 12: tiny 5-token transformer block per batch element.  block=160.
// ---------------------------------------------------------------------------
__global__ void k_xformer(const float* __restrict__ aiw, const float* __restrict__ aib,
                          const float* __restrict__ aow, const float* __restrict__ aob,
                          const float* __restrict__ g1, const float* __restrict__ bb1,
                          const float* __restrict__ g2, const float* __restrict__ bb2,
                          const float* __restrict__ f1w, const float* __restrict__ f1b,
                          const float* __restrict__ f2w, const float* __restrict__ f2b,
                          const float* __restrict__ seq_in, float* __restrict__ h2cat) {
  int b = blockIdx.x, t = threadIdx.x;
  __shared__ float s[5][14], q[5][14], kk[5][14], v[5][14];
  __shared__ float att[5][5], tmp[5][14], ff1[5][28];
  __shared__ float rmean[5], rstd[5];
  int r = t / 14, c = t % 14;
  if (t < 70) s[r][c] = seq_in[b * 70 + t];
  __syncthreads();
  if (t < 70) {  // qkv projection
    float sq = aib[c], sk = aib[14 + c], sv = aib[28 + c];
    for (int j = 0; j < 14; ++j) {
      float x = s[r][j];
      sq += x * aiw[c * 14 + j];
      sk += x * aiw[(14 + c) * 14 + j];
      sv += x * aiw[(28 + c) * 14 + j];
    }
    q[r][c] = sq; kk[r][c] = sk; v[r][c] = sv;
  }
  __syncthreads();
  if (t < 25) {  // scores
    int rr = t / 5, cc = t % 5;
    float e = 0.f;
    for (int j = 0; j < 14; ++j) e += q[rr][j] * kk[cc][j];
    att[rr][cc] = e * rsqrtf(14.f);
  }
  __syncthreads();
  if (t < 5) {  // softmax per row
    float mx = att[t][0];
    for (int j = 1; j < 5; ++j) mx = fmaxf(mx, att[t][j]);
    float sm = 0.f;
    for (int j = 0; j < 5; ++j) { float e = __expf(att[t][j] - mx); att[t][j] = e; sm += e; }
    float inv = 1.f / sm;
    for (int j = 0; j < 5; ++j) att[t][j] *= inv;
  }
  __syncthreads();
  if (t < 70) {  // attn @ v
    float e = 0.f;
    for (int j = 0; j < 5; ++j) e += att[r][j] * v[j][c];
    tmp[r][c] = e;
  }
  __syncthreads();
  if (t < 70) {  // out proj + residual -> q
    float e = aob[c];
    for (int j = 0; j < 14; ++j) e += tmp[r][j] * aow[c * 14 + j];
    q[r][c] = s[r][c] + e;
  }
  __syncthreads();
  if (t < 5) {  // LN1 stats
    float m = 0.f;
    for (int j = 0; j < 14; ++j) m += q[t][j];
    m *= (1.f / 14.f);
    float var = 0.f;
    for (int j = 0; j < 14; ++j) { float d = q[t][j] - m; var += d * d; }
    var *= (1.f / 14.f);
    rmean[t] = m; rstd[t] = rsqrtf(var + 1e-5f);
  }
  __syncthreads();
  if (t < 70) s[r][c] = (q[r][c] - rmean[r]) * rstd[r] * g1[c] + bb1[c];
  __syncthreads();
  if (t < 140) {  // ffn1 + relu
    int rr = t / 28, o = t % 28;
    float e = f1b[o];
    for (int j = 0; j < 14; ++j) e += s[rr][j] * f1w[o * 14 + j];
    ff1[rr][o] = fmaxf(e, 0.f);
  }
  __syncthreads();
  if (t < 70) {  // ffn2 + residual -> q
    float e = f2b[c];
    for (int j = 0; j < 28; ++j) e += ff1[r][j] * f2w[c * 28 + j];
    q[r][c] = s[r][c] + e;
  }
  __syncthreads();
  if (t < 5) {  // LN2 stats
    float m = 0.f;
    for (int j = 0; j < 14; ++j) m += q[t][j];
    m *= (1.f / 14.f);
    float var = 0.f;
    for (int j = 0; j < 14; ++j) { float d = q[t][j] - m; var += d * d; }
    var *= (1.f / 14.f);
    rmean[t] = m; rstd[t] = rsqrtf(var + 1e-5f);
  }
  __syncthreads();
  if (t < 70) s[r][c] = (q[r][c] - rmean[r]) * rstd[r] * g2[c] + bb2[c];
  __syncthreads();
  if (t < 28) {
    if (t < 14) h2cat[b * 28 + t] = s[0][t];
    else {
      int cc = t - 14;
      h2cat[b * 28 + t] = (s[1][cc] + s[2][cc] + s[3][cc] + s[4][cc]) * 0.25f;
    }
  }
}

// ---------------------------------------------------------------------------
// Stage 13: MLP head 28 -> 512 -> 128 -> 6.  grid=64, block=512.
// ---------------------------------------------------------------------------
__global__ void k_head(const float* __restrict__ h2cat,
                       const float* __restrict__ w1, const float* __restrict__ b1,
                       const float* __restrict__ w2, const float* __restrict__ b2,
                       const float* __restrict__ w3, const float* __restrict__ b3,
                       float* __restrict__ out) {
  int b = blockIdx.x, t = threadIdx.x;
  __shared__ float a0[28], a1[512], a2[128];
  if (t < 28) a0[t] = h2cat[b * 28 + t];
  __syncthreads();
  {
    float e = b1[t];
    for (int j = 0; j < 28; ++j) e += a0[j] * w1[t * 28 + j];
    a1[t] = fmaxf(e, 0.f);
  }
  __syncthreads();
  if (t < 128) {
    float e = b2[t];
    for (int j = 0; j < 512; ++j) e += a1[j] * w2[t * 512 + j];
    a2[t] = fmaxf(e, 0.f);
  }
  __syncthreads();
  if (t < 6) {
    float e = b3[t];
    for (int j = 0; j < 128; ++j) e += a2[j] * w3[t * 128 + j];
    out[b * 6 + t] = e;
  }
}

// ---------------------------------------------------------------------------
extern "C" void kernel_launch(void* const* d_in, const int* in_sizes, int n_in,
                              void* d_out, int out_size, void* d_ws, size_t ws_size,
                              hipStream_t stream) {
  const float* x        = (const float*)d_in[0];
  const float* conv1_w  = (const float*)d_in[1];
  const float* conv1_b  = (const float*)d_in[2];
  const float* conv2_w  = (const float*)d_in[3];
  const float* conv2_b  = (const float*)d_in[4];
  const float* conv3_w  = (const float*)d_in[5];
  const float* conv3_b  = (const float*)d_in[6];
  const float* venc_w   = (const float*)d_in[7];
  const float* venc_b   = (const float*)d_in[8];
  const float* tenc1_w  = (const float*)d_in[9];
  const float* tenc1_b  = (const float*)d_in[10];
  const float* tenc2_w  = (const float*)d_in[11];
  const float* tenc2_b  = (const float*)d_in[12];
  const float* attn_in_w  = (const float*)d_in[13];
  const float* attn_in_b  = (const float*)d_in[14];
  const float* attn_out_w = (const float*)d_in[15];
  const float* attn_out_b = (const float*)d_in[16];
  const float* ln1_g = (const float*)d_in[17];
  const float* ln1_b = (const float*)d_in[18];
  const float* ln2_g = (const float*)d_in[19];
  const float* ln2_b = (const float*)d_in[20];
  const float* ffn1_w = (const float*)d_in[21];
  const float* ffn1_b = (const float*)d_in[22];
  const float* ffn2_w = (const float*)d_in[23];
  const float* ffn2_b = (const float*)d_in[24];
  const float* head1_w = (const float*)d_in[25];
  const float* head1_b = (const float*)d_in[26];
  const float* head2_w = (const float*)d_in[27];
  const float* head2_b = (const float*)d_in[28];
  const float* head3_w = (const float*)d_in[29];
  const float* head3_b = (const float*)d_in[30];
  const float* mem_nodes = (const float*)d_in[31];
  const float* mem_edges = (const float*)d_in[32];
  float* out = (float*)d_out;

  // workspace layout
  char* w = (char*)d_ws;
  float* h1     = (float*)w; w += (size_t)64 * 16 * 961 * 4;
  float* c2full = (float*)w; w += (size_t)64 * 32 * 961 * 4;
  float* h2     = (float*)w; w += (size_t)64 * 32 * 225 * 4;
  float* h3     = (float*)w; w += (size_t)64 * 32 * 225 * 4;
  float* cost   = (float*)w; w += (size_t)64 * 256 * 256 * 4;
  float* ve     = (float*)w; w += (size_t)64 * 8 * 4;
  int*   cl     = (int*)w;   w += (size_t)64 * 4;
  float* dfin   = (float*)w; w += (size_t)64 * 256 * 4;
  int*   topk   = (int*)w;   w += (size_t)64 * 4 * 4;
  float* seq    = (float*)w; w += (size_t)64 * 70 * 4;
  float* h2cat  = (float*)w; w += (size_t)64 * 28 * 4;

  k_tenc<<<64, 64, 0, stream>>>(x, tenc1_w, tenc1_b, tenc2_w, tenc2_b, seq);
  k_conv1<<<3844, 256, 0, stream>>>(x, conv1_w, conv1_b, h1);
  k_conv2<<<dim3(61, 64), 32, 0, stream>>>(h1, conv2_w, conv2_b, c2full);
  k_pool2<<<1800, 256, 0, stream>>>(c2full, h2);
  k_conv3<<<dim3(15, 64), 32, 0, stream>>>(h2, conv3_w, conv3_b, h3);
  k_venc<<<dim3(8, 64), 256, 0, stream>>>(h3, venc_w, venc_b, ve);
  k_closest<<<64, 256, 0, stream>>>(ve, mem_nodes, cl);
  k_cost<<<16384, 256, 0, stream>>>(mem_edges, cost);
  k_bellman<<<64, 256, 0, stream>>>(cost, cl, dfin);
  k_topk<<<64, 256, 0, stream>>>(dfin, topk);
  k_gather<<<dim3(4, 64), 256, 0, stream>>>(mem_nodes, mem_edges, topk, seq);
  k_xformer<<<64, 160, 0, stream>>>(attn_in_w, attn_in_b, attn_out_w, attn_out_b,
                                    ln1_g, ln1_b, ln2_g, ln2_b,
                                    ffn1_w, ffn1_b, ffn2_w, ffn2_b, seq, h2cat);
  k_head<<<64, 512, 0, stream>>>(h2cat, head1_w, head1_b, head2_w, head2_b,
                                 head3_w, head3_b, out);
}